// SubjectBatchNormA_54004918780586
// MI455X (gfx1250) — compile-verified
//
#include <hip/hip_runtime.h>
#include <stdint.h>

typedef float v4f __attribute__((ext_vector_type(4)));

#define C_   128
#define B_   64
#define S_   4
#define HW_  4096          // 64*64 spatial
#define N_   262144.0f     // B*H*W, BN normalizer (full batch incl. zeros)
#define EPS_ 1e-5f

// ---------------------------------------------------------------------------
// Pass 1: per-(b,c) plane sums. One wave (32 lanes) owns one 4096-float plane:
// 32 coalesced b128 loads per lane, then a wave32 shfl_xor tree reduction.
// 8 waves / block -> 1024 blocks, no LDS, no barriers.
// ---------------------------------------------------------------------------
__global__ __launch_bounds__(256) void sbn_reduce(const float* __restrict__ x,
                                                  float* __restrict__ xs,
                                                  float* __restrict__ xs2) {
  const int lane  = threadIdx.x & 31;
  const int wave  = threadIdx.x >> 5;
  const int plane = blockIdx.x * 8 + wave;           // bc = b*C + c
  const v4f* p = (const v4f*)(x + (size_t)plane * HW_);
  float s1 = 0.f, s2 = 0.f;
#pragma unroll 8
  for (int k = 0; k < 32; ++k) {
    v4f v = p[k * 32 + lane];
    s1 += (v.x + v.y) + (v.z + v.w);
    s2 = fmaf(v.x, v.x, s2);
    s2 = fmaf(v.y, v.y, s2);
    s2 = fmaf(v.z, v.z, s2);
    s2 = fmaf(v.w, v.w, s2);
  }
#pragma unroll
  for (int off = 16; off > 0; off >>= 1) {
    s1 += __shfl_xor(s1, off, 32);
    s2 += __shfl_xor(s2, off, 32);
  }
  if (lane == 0) { xs[plane] = s1; xs2[plane] = s2; }
}

// ---------------------------------------------------------------------------
// Pass 2: one tiny block (128 threads = one channel each). Stage xs/xs2
// (64 KiB) into LDS with CDNA5 async global->LDS copies (ASYNCcnt), then
// fold BN stats into per-(s,c) scale/bias, including the cross-subject
// z_total - z_s term.
// ---------------------------------------------------------------------------
__global__ __launch_bounds__(128) void sbn_stats(const int*   __restrict__ subj,
                                                 const float* __restrict__ gamma,
                                                 const float* __restrict__ beta,
                                                 const float* __restrict__ xs,
                                                 const float* __restrict__ xs2,
                                                 float* __restrict__ scale,
                                                 float* __restrict__ bias) {
  __shared__ float lds[2 * B_ * C_];                 // [0..8191]=xs, [8192..]=xs2
  const int t = threadIdx.x;                         // channel c
  const uint32_t lbase = (uint32_t)(uintptr_t)(&lds[0]);

  // 2048 float4 per table, 128 threads -> 16 async b128 copies each, x2 tables
#pragma unroll
  for (int k = 0; k < 16; ++k) {
    const int i4 = k * 128 + t;
    const float* g0 = xs + (size_t)i4 * 4;
    const float* g1 = xs2 + (size_t)i4 * 4;
    uint32_t l0 = lbase + (uint32_t)i4 * 16u;
    uint32_t l1 = l0 + 32768u;
    asm volatile("global_load_async_to_lds_b128 %0, %1, off" :: "v"(l0), "v"(g0) : "memory");
    asm volatile("global_load_async_to_lds_b128 %0, %1, off" :: "v"(l1), "v"(g1) : "memory");
  }
  asm volatile("s_wait_asynccnt 0x0" ::: "memory");  // this wave's copies done
  __syncthreads();                                   // all waves' copies visible

  const int c = t;
  float mean[S_], iv[S_], zz[S_];
  float ztot = 0.f;
#pragma unroll
  for (int s = 0; s < S_; ++s) {
    float s1 = 0.f, s2 = 0.f;
    int cnt = 0;
    for (int b = 0; b < B_; ++b) {
      if (subj[b] == s) {
        s1 += lds[b * C_ + c];
        s2 += lds[B_ * C_ + b * C_ + c];
        ++cnt;
      }
    }
    const float m  = s1 / N_;
    const float ve = s2 / N_ - m * m + EPS_;
    const float r  = rsqrtf(ve);
    const float g  = gamma[s * C_ + c];
    const float be = beta[s * C_ + c];
    const float z  = (cnt > 0) ? fmaf(-m * r, g, be) : 0.f;  // BN value at zeroed pos
    mean[s] = m; iv[s] = r; zz[s] = z;
    ztot += z;
  }
#pragma unroll
  for (int s = 0; s < S_; ++s) {
    const float g  = gamma[s * C_ + c];
    const float sc = iv[s] * g;
    scale[s * C_ + c] = sc;
    bias[s * C_ + c]  = beta[s * C_ + c] - mean[s] * sc + (ztot - zz[s]);
  }
}

// ---------------------------------------------------------------------------
// Pass 3: out = x*scale[s,c] + bias[s,c]. Each block covers 256 float4 =
// exactly 1/4 of one (b,c) plane, so s/scale/bias are blockIdx-uniform
// (scalarized to SMEM loads). NT store keeps x resident in the 192MB L2.
// ---------------------------------------------------------------------------
__global__ __launch_bounds__(256) void sbn_apply(const float* __restrict__ x,
                                                 const int*   __restrict__ subj,
                                                 const float* __restrict__ scale,
                                                 const float* __restrict__ bias,
                                                 float* __restrict__ out) {
  const int bc = blockIdx.x >> 2;                    // 4 blocks per plane
  const int b  = bc >> 7;
  const int c  = bc & (C_ - 1);
  const int s  = subj[b];
  const float sc = scale[s * C_ + c];
  const float bi = bias[s * C_ + c];

  const size_t i = (size_t)blockIdx.x * 256 + threadIdx.x;
  v4f v = ((const v4f*)x)[i];
  v4f o;
  o.x = fmaf(v.x, sc, bi);
  o.y = fmaf(v.y, sc, bi);
  o.z = fmaf(v.z, sc, bi);
  o.w = fmaf(v.w, sc, bi);
  __builtin_nontemporal_store(o, (v4f*)out + i);
}

extern "C" void kernel_launch(void* const* d_in, const int* in_sizes, int n_in,
                              void* d_out, int out_size, void* d_ws, size_t ws_size,
                              hipStream_t stream) {
  const float* x     = (const float*)d_in[0];
  const int*   subj  = (const int*)d_in[1];
  const float* gamma = (const float*)d_in[2];
  const float* beta  = (const float*)d_in[3];
  float* out = (float*)d_out;

  // workspace layout (floats): xs[8192] | xs2[8192] | scale[512] | bias[512]
  float* xs    = (float*)d_ws;
  float* xs2   = xs + B_ * C_;
  float* scale = xs2 + B_ * C_;
  float* bias  = scale + S_ * C_;

  sbn_reduce<<<(B_ * C_) / 8, 256, 0, stream>>>(x, xs, xs2);
  sbn_stats<<<1, C_, 0, stream>>>(subj, gamma, beta, xs, xs2, scale, bias);

  const int total4 = (B_ * C_ * HW_) / 4;            // 8,388,608 float4
  sbn_apply<<<total4 / 256, 256, 0, stream>>>(x, subj, scale, bias, out);
}